// DnlsLoss_16621523435653
// MI455X (gfx1250) — compile-verified
//
#include <hip/hip_runtime.h>
#include <hip/hip_bf16.h>

// ---------------- problem constants (from reference) ----------------
#define T_       5
#define C_       3
#define H_       384
#define W_       384
#define PS_      7
#define WS_      9
#define WT_      1
#define K_       10
#define STRIDE0_ 4
#define NH_      (H_/STRIDE0_)      // 96
#define NW_      (W_/STRIDE0_)      // 96
#define NQ_      (T_*NH_*NW_)       // 46080
#define PE_      (PS_*PS_*C_)       // 147 elements per patch
#define NT_      (2*WT_+1)          // 3 temporal offsets
#define NCAND_   (NT_*WS_*WS_)      // 243 candidates per query
#define NGROUPS_ 16                 // 16 groups x 16 candidates = 256 (padded)
#define REG_     15                 // candidate region side: (WS-1)+(PS-1)+1 = 15
#define REGSZ_   (REG_*REG_*C_)     // 675 floats per region
#define BIGF_    3.0e38f

typedef float v2f __attribute__((ext_vector_type(2)));
typedef float v8f __attribute__((ext_vector_type(8)));

// Region-relative offset of patch element e (e = dy*21 + dx*3 + ch):
//   off = dy*45 + dx*3 + ch = e + 24*(e/21).  Pure compile-time per unrolled chunk.
__host__ __device__ constexpr int offt(int e) { return e + 24 * (e / 21); }

// CDNA5 async gather: one f32 from global memory directly into LDS.
// LDS offset = low 32 bits of the generic LDS pointer (aperture low bits == DS offset).
__device__ __forceinline__ void async_copy_f32(float* dst_lds, const float* src_glb) {
  unsigned           lds = (unsigned)(size_t)dst_lds;
  unsigned long long ga  = (unsigned long long)(size_t)src_glb;
  asm volatile("global_load_async_to_lds_b32 %0, %1, off"
               :: "v"(lds), "v"(ga) : "memory");
}
__device__ __forceinline__ void async_wait0() {
  asm volatile("s_wait_asynccnt 0" ::: "memory");
}

// One workgroup (one wave32) per query.
__global__ __launch_bounds__(32) void nls_topk_kernel(
    const float* __restrict__ noisy,
    const float* __restrict__ fflow,
    const float* __restrict__ bflow,
    float* __restrict__ qsums)
{
  const int qi   = blockIdx.x;          // query index
  const int lane = threadIdx.x;         // 0..31
  const int t    = qi / (NH_*NW_);
  const int rq   = qi % (NH_*NW_);
  const int hi   = (rq / NW_) * STRIDE0_;
  const int wi   = (rq % NW_) * STRIDE0_;

  __shared__ float sq[PE_ + 1];                 // query patch (+1 zero pad slot)
  __shared__ float sreg[NT_][REGSZ_];           // candidate regions per dt
  __shared__ float sdist[NGROUPS_ * 16];        // 256 candidate distances
  __shared__ int   sp_tj[NT_], sp_bh[NT_], sp_bw[NT_], sp_r0[NT_], sp_c0[NT_];

  // ---- per-dt anchors from flow (lanes 0..2) ----
  if (lane < NT_) {
    const int di = lane;                        // 0:bflow(dt=-1) 1:zero 2:fflow(dt=+1)
    int tj = t + di - WT_;
    tj = tj < 0 ? 0 : (tj > T_-1 ? T_-1 : tj);
    float fx = 0.f, fy = 0.f;
    if (di == 0) {
      fx = bflow[((size_t)(t*2 + 0)*H_ + hi)*W_ + wi];
      fy = bflow[((size_t)(t*2 + 1)*H_ + hi)*W_ + wi];
    } else if (di == 2) {
      fx = fflow[((size_t)(t*2 + 0)*H_ + hi)*W_ + wi];
      fy = fflow[((size_t)(t*2 + 1)*H_ + hi)*W_ + wi];
    }
    int bh = hi + (int)rintf(fy);               // jnp.round == RNE == rintf
    int bw = wi + (int)rintf(fx);
    bh = bh < 0 ? 0 : (bh > H_-1 ? H_-1 : bh);
    bw = bw < 0 ? 0 : (bw > W_-1 ? W_-1 : bw);
    sp_tj[di] = tj; sp_bh[di] = bh; sp_bw[di] = bw;
    sp_r0[di] = (bh - WS_/2 < 0) ? 0 : bh - WS_/2;
    sp_c0[di] = (bw - WS_/2 < 0) ? 0 : bw - WS_/2;
  }
  if (lane == 0) sq[PE_] = 0.f;                 // zero pad for last-chunk high half
  __syncthreads();

  // ---- stage query patch into LDS (async gather, edge-clamped) ----
  for (int e = lane; e < PE_; e += 32) {
    int dy = e / (PS_*C_);
    int rm = e - dy*(PS_*C_);
    int dx = rm / C_;
    int ch = rm - dx*C_;
    int row = hi + dy; row = row > H_-1 ? H_-1 : row;
    int col = wi + dx; col = col > W_-1 ? W_-1 : col;
    async_copy_f32(&sq[e], &noisy[(((size_t)t*C_ + ch)*H_ + row)*W_ + col]);
  }
  // ---- stage candidate regions; edge rows/cols replicated (this IS the clamp) ----
  for (int di = 0; di < NT_; ++di) {
    const int tj = sp_tj[di], r0 = sp_r0[di], c0 = sp_c0[di];
    for (int idx = lane; idx < REGSZ_; idx += 32) {
      int rr = idx / (REG_*C_);
      int rm = idx - rr*(REG_*C_);
      int cc = rm / C_;
      int ch = rm - cc*C_;
      int arow = r0 + rr; arow = arow > H_-1 ? H_-1 : arow;
      int acol = c0 + cc; acol = acol > W_-1 ? W_-1 : acol;
      async_copy_f32(&sreg[di][idx],
                     &noisy[(((size_t)tj*C_ + ch)*H_ + arow)*W_ + acol]);
    }
  }
  async_wait0();
  __syncthreads();

  // ---- WMMA Gram-diagonal distance engine ----
  // A (16x4 f32): lane L holds candidate m = L%16; VGPR0/1 = diff at K = k0/k0+1,
  // k0 = 0 (lanes<16) or 2 (lanes>=16).  B = A^T has the identical per-lane register
  // image for a symmetric Gram product, so the same v2f feeds both operands.
  const int  m       = lane & 15;
  const bool lowhalf = (lane < 16);
  const float* qb    = &sq[lowhalf ? 0 : 2];    // per-half-wave query base

  for (int g = 0; g < NGROUPS_; ++g) {
    int ci  = g*16 + m;
    int cic = (ci < NCAND_) ? ci : (NCAND_-1);  // clamp pads to a valid candidate
    int di  = cic / (WS_*WS_);
    int s   = cic - di*(WS_*WS_);
    int sy  = s / WS_ - WS_/2;
    int sx  = (s - (s/WS_)*WS_) - WS_/2;
    int hj  = sp_bh[di] + sy; hj = hj < 0 ? 0 : (hj > H_-1 ? H_-1 : hj);
    int wj  = sp_bw[di] + sx; wj = wj < 0 ? 0 : (wj > W_-1 ? W_-1 : wj);
    // affine candidate base; per-element clamp is absorbed by region replication
    const int sbase = ((hj - sp_r0[di])*REG_ + (wj - sp_c0[di]))*C_;  // <= 384
    const float* regb = &sreg[di][sbase];

    v8f acc = {};
    #pragma unroll
    for (int c = 0; c < 36; ++c) {              // chunks 0..35 fully in-range
      const int e0 = 4*c;
      float q0 = qb[e0];                        // sq[e0 + (lowhalf?0:2)]
      float q1 = qb[e0 + 1];
      int  oA  = lowhalf ? offt(e0)     : offt(e0 + 2);   // cndmask of constants
      int  oB  = lowhalf ? offt(e0 + 1) : offt(e0 + 3);
      float c0v = regb[oA];
      float c1v = regb[oB];
      v2f a; a.x = q0 - c0v; a.y = q1 - c1v;
      acc = __builtin_amdgcn_wmma_f32_16x16x4_f32(
          false, a, false, a, (short)0, acc, false, false);
    }
    { // chunk 36: elements 144..146 valid; 147 is zero pad (high half, 2nd slot)
      float q0 = qb[144];
      float q1 = qb[145];                       // == sq[147] == 0 for high half
      int  oA  = lowhalf ? offt(144) : offt(146);
      int  oB  = lowhalf ? offt(145) : 0;
      float c0v = regb[oA];
      float c1v = regb[oB];
      v2f a;
      a.x = q0 - c0v;
      a.y = lowhalf ? (q1 - c1v) : 0.f;
      acc = __builtin_amdgcn_wmma_f32_16x16x4_f32(
          false, a, false, a, (short)0, acc, false, false);
    }

    // Extract diagonal D[i][i]:
    //   i in 0..7  -> lane i,    acc[i]
    //   i in 8..15 -> lane 16+i, acc[i-8]   (lanes 24..31)
    const int want = (lane < 8) ? lane : ((lane >= 24) ? (lane - 24) : -1);
    float dval = 0.f;
    #pragma unroll
    for (int rsel = 0; rsel < 8; ++rsel)
      if (want == rsel) dval = acc[rsel];
    if (want >= 0) {
      const int ii   = (lane < 8) ? lane : (lane - 16);
      const int cidx = g*16 + ii;
      sdist[cidx] = (cidx < NCAND_) ? dval : BIGF_;  // pads excluded from top-K
    }
  }
  __syncthreads();

  // ---- iterative top-K (K=10 smallest of 256) ----
  float total = 0.f;
  for (int k = 0; k < K_; ++k) {
    float bv = BIGF_; int bi = 0x7fffffff;
    #pragma unroll
    for (int j = 0; j < 8; ++j) {
      const int idx = lane + j*32;
      const float x = sdist[idx];
      if (x < bv) { bv = x; bi = idx; }
    }
    for (int off = 16; off > 0; off >>= 1) {
      const float ov = __shfl_xor(bv, off, 32);
      const int   oi = __shfl_xor(bi, off, 32);
      if (ov < bv || (ov == bv && oi < bi)) { bv = ov; bi = oi; }
    }
    total += bv;                 // uniform across the wave
    __syncthreads();
    if (lane == 0) sdist[bi] = BIGF_;
    __syncthreads();
  }
  if (lane == 0) qsums[qi] = total;
}

// Deterministic fixed-order reduction: mean over NQ_*K_ distances.
__global__ __launch_bounds__(256) void mean_reduce_kernel(
    const float* __restrict__ qsums, float* __restrict__ out)
{
  __shared__ double sbuf[256];
  const int tid = threadIdx.x;
  double s = 0.0;
  for (int i = tid; i < NQ_; i += 256) s += (double)qsums[i];
  sbuf[tid] = s;
  __syncthreads();
  for (int off = 128; off > 0; off >>= 1) {
    if (tid < off) sbuf[tid] += sbuf[tid + off];
    __syncthreads();
  }
  if (tid == 0) out[0] = (float)(sbuf[0] / (double)((long long)NQ_ * K_));
}

extern "C" void kernel_launch(void* const* d_in, const int* in_sizes, int n_in,
                              void* d_out, int out_size, void* d_ws, size_t ws_size,
                              hipStream_t stream) {
  const float* noisy = (const float*)d_in[0];
  // d_in[1] = deno (unused: SEARCH_INPUT == 'noisy' and refine uses noisy)
  const float* fflow = (const float*)d_in[2];
  const float* bflow = (const float*)d_in[3];
  float* out   = (float*)d_out;
  float* qsums = (float*)d_ws;              // NQ_ floats of scratch

  nls_topk_kernel<<<NQ_, 32, 0, stream>>>(noisy, fflow, bflow, qsums);
  mean_reduce_kernel<<<1, 256, 0, stream>>>(qsums, out);
}